// SparseCodebookEmb_33105607918086
// MI455X (gfx1250) — compile-verified
//
#include <hip/hip_runtime.h>

#ifndef __has_builtin
#define __has_builtin(x) 0
#endif

// ---- constants from the reference ----
#define NUM_FIELDS 26
#define HIDDEN     64
#define MAX_NNZ    16
#define PAIRS_PER_BLOCK 16   // 256 threads, 16 lanes per (b,f) pair
#define CB_V4      (NUM_FIELDS * (HIDDEN / 4))   // 416 float4 = 6656 B codebook

typedef float v4f __attribute__((ext_vector_type(4)));
typedef int   v4i __attribute__((ext_vector_type(4)));
// may_alias so the float4 base store / scalar scatter / float4 readback in LDS
// cannot be reordered or store-forwarded past each other by TBAA.
typedef v4f   __attribute__((may_alias)) v4f_ma;
typedef float __attribute__((may_alias)) f_ma;

#if __has_builtin(__builtin_amdgcn_global_load_async_to_lds_b128)
#define HAVE_ASYNC_LDS 1
// Builtin signature (from hipcc diagnostic): (v4i addrspace(1)*, v4i addrspace(3)*, Imm i32, Imm i32)
typedef __attribute__((address_space(1))) v4i gas_v4i;
typedef __attribute__((address_space(3))) v4i las_v4i;
#endif

__global__ __launch_bounds__(256)
void SparseCodebookEmb_kernel(const int*   __restrict__ x,        // [B*F]
                              const float* __restrict__ codebook, // [F,64]
                              const float* __restrict__ values,   // [NumFeat,16]
                              const int*   __restrict__ cols,     // [NumFeat,16]
                              const int*   __restrict__ nnz,      // [NumFeat]
                              float*       __restrict__ out,      // [B*F,64]
                              int n_pairs) {
    __shared__ v4f cbuf[CB_V4];                              // staged codebook
    __shared__ v4f rowbuf[PAIRS_PER_BLOCK * (HIDDEN / 4)];   // 16 x 256B rows

    const int tid = threadIdx.x;

    // ---- stage codebook into LDS (all threads participate, then barrier) ----
#if HAVE_ASYNC_LDS
    for (int i = tid; i < CB_V4; i += blockDim.x) {
        __builtin_amdgcn_global_load_async_to_lds_b128(
            (gas_v4i*)(codebook + (size_t)i * 4),
            (las_v4i*)&cbuf[i], 0, 0);
    }
#if __has_builtin(__builtin_amdgcn_s_wait_asynccnt)
    __builtin_amdgcn_s_wait_asynccnt(0);
#else
    asm volatile("s_wait_asynccnt 0" ::: "memory");
#endif
#else
    for (int i = tid; i < CB_V4; i += blockDim.x) {
        cbuf[i] = ((const v4f*)codebook)[i];   // global_load_b128 + ds_store_b128
    }
#endif
    __syncthreads();

    const int half = tid >> 4;   // which pair slot in this block (0..15)
    const int l    = tid & 15;   // lane within the pair (0..15), owns float4 #l
    const int p    = blockIdx.x * PAIRS_PER_BLOCK + half;

    if (p < n_pairs) {
        const int f  = p % NUM_FIELDS;
        const int id = x[p];

        // kick the CSR row gathers early (global_prefetch_b8)
        __builtin_prefetch(&values[id * MAX_NNZ], 0, 3);
        __builtin_prefetch(&cols[id * MAX_NNZ],   0, 3);

        const int   nn   = nnz[id];
        const int   colv = cols[id * MAX_NNZ + l] & (HIDDEN - 1); // safety clamp
        const float valv = values[id * MAX_NNZ + l];

        // base row chunk from LDS-staged codebook (ds_load_b128)
        const v4f base = cbuf[f * (HIDDEN / 4) + l];

        v4f_ma* rb4 = (v4f_ma*)&rowbuf[half * (HIDDEN / 4)];
        f_ma*   rbf = (f_ma*)rb4;

        // LDS ops from one wave are in-order (DScnt): store -> scatter -> load.
        rb4[l] = base;                      // ds_store_b128 : 64-float base row
        if (l < nn) rbf[colv] = valv;       // ds_store_b32  : cols unique per row
        const v4f outv = rb4[l];            // ds_load_b128  : assembled row

        ((v4f*)out)[(size_t)p * (HIDDEN / 4) + l] = outv;  // global_store_b128
    }
}

extern "C" void kernel_launch(void* const* d_in, const int* in_sizes, int n_in,
                              void* d_out, int out_size, void* d_ws, size_t ws_size,
                              hipStream_t stream) {
    const int*   x        = (const int*)  d_in[0];
    const float* codebook = (const float*)d_in[1];
    const float* values   = (const float*)d_in[2];
    const int*   cols     = (const int*)  d_in[3];
    const int*   nnz      = (const int*)  d_in[4];
    float*       out      = (float*)d_out;

    const int n_pairs = out_size / HIDDEN;                       // 4096*26
    const int blocks  = (n_pairs + PAIRS_PER_BLOCK - 1) / PAIRS_PER_BLOCK;

    hipLaunchKernelGGL(SparseCodebookEmb_kernel, dim3(blocks), dim3(256), 0, stream,
                       x, codebook, values, cols, nnz, out, n_pairs);
}